// Generator1_9208409883011
// MI455X (gfx1250) — compile-verified
//
#include <hip/hip_runtime.h>
#include <math.h>

typedef __attribute__((ext_vector_type(2))) float v2f;
typedef __attribute__((ext_vector_type(8))) float v8f;

#define BN_EPS 1e-3f
#define D1 35
#define D2 160
#define EPB 32   // edges per block in k_edge2

// ---------------------------------------------------------------- zero scratch
__global__ void k_zero(float* __restrict__ p, int n) {
  int i = blockIdx.x * blockDim.x + threadIdx.x;
  if (i < n) p[i] = 0.0f;
}

// ------------------------------------------------- layer-1 edge messages (35ch)
// msg[e,o] = sum_i x[src_e,i] * relu(a_e * w1[i*35+o] + b1[i*35+o])
__global__ void k_edge1(const float* __restrict__ x,
                        const long long* __restrict__ ei,   // [2,E] int64
                        const float* __restrict__ ea,
                        const float* __restrict__ w1,
                        const float* __restrict__ b1,
                        float* __restrict__ agg1,
                        float* __restrict__ cnt,
                        int E) {
  int idx = blockIdx.x * blockDim.x + threadIdx.x;
  int total = E * D1;
  if (idx >= total) return;
  int e = idx / D1;
  int o = idx - e * D1;
  int src = (int)ei[e];
  int dst = (int)ei[E + e];
  float a = ea[e];
  const float* xr = x + (size_t)src * D1;
  float s = 0.0f;
#pragma unroll
  for (int i = 0; i < D1; ++i) {
    float w = fmaf(a, w1[i * D1 + o], b1[i * D1 + o]);
    w = fmaxf(w, 0.0f);
    s = fmaf(xr[i], w, s);
  }
  atomicAdd(&agg1[(size_t)dst * D1 + o], s);
  if (o == 0) atomicAdd(&cnt[dst], 1.0f);
}

// --------------------------------------- combine1: mean + root + BN + sigmoid
__global__ void k_combine1(const float* __restrict__ x,
                           const float* __restrict__ agg1,
                           const float* __restrict__ cnt,
                           const float* __restrict__ root1,  // [35,35]
                           const float* __restrict__ bias1,
                           const float* __restrict__ g,
                           const float* __restrict__ be,
                           const float* __restrict__ mu,
                           const float* __restrict__ var,
                           float* __restrict__ x1,
                           int N) {
  int idx = blockIdx.x * blockDim.x + threadIdx.x;
  int total = N * D1;
  if (idx >= total) return;
  int n = idx / D1;
  int o = idx - n * D1;
  float c = fmaxf(cnt[n], 1.0f);
  float h = agg1[idx] / c;
  const float* xr = x + (size_t)n * D1;
#pragma unroll
  for (int i = 0; i < D1; ++i) h = fmaf(xr[i], root1[i * D1 + o], h);
  h += bias1[o];
  h = (h - mu[o]) * rsqrtf(var[o] + BN_EPS) * g[o] + be[o];
  x1[idx] = 1.0f / (1.0f + expf(-h));
}

// ------------------------------------------------ layer-2 edge messages (160ch)
// thread = output channel o (blockDim=160, 5 waves); block owns EPB edges.
// Per-channel weight columns live in 70 registers (amortized over EPB edges);
// the 35-float x1[src] row is staged through LDS once per edge.
__global__ void __launch_bounds__(D2)
k_edge2(const float* __restrict__ x1,
        const long long* __restrict__ ei,
        const float* __restrict__ ea,
        const float* __restrict__ w2,   // [35*160]
        const float* __restrict__ b2,
        float* __restrict__ agg2,
        int E) {
  const int o = threadIdx.x;            // 0..159
  const int e0 = blockIdx.x * EPB;
  const int e1 = (e0 + EPB < E) ? (e0 + EPB) : E;

  float wc[D1], bc[D1];
#pragma unroll
  for (int i = 0; i < D1; ++i) {
    wc[i] = w2[i * D2 + o];
    bc[i] = b2[i * D2 + o];
  }

  __shared__ float xr[D1];
  for (int e = e0; e < e1; ++e) {
    int src = (int)ei[e];
    int dst = (int)ei[E + e];
    float a = ea[e];
    __syncthreads();
    if (threadIdx.x < D1) xr[threadIdx.x] = x1[(size_t)src * D1 + threadIdx.x];
    __syncthreads();
    float s = 0.0f;
#pragma unroll
    for (int i = 0; i < D1; ++i) {
      float w = fmaf(a, wc[i], bc[i]);
      s = fmaf(xr[i], fmaxf(w, 0.0f), s);
    }
    atomicAdd(&agg2[(size_t)dst * D2 + o], s);
  }
}

// --------------------------------------- combine2: mean + root + BN + sigmoid
__global__ void k_combine2(const float* __restrict__ x1,
                           const float* __restrict__ agg2,
                           const float* __restrict__ cnt,
                           const float* __restrict__ root2,  // [35,160]
                           const float* __restrict__ bias2,
                           const float* __restrict__ g,
                           const float* __restrict__ be,
                           const float* __restrict__ mu,
                           const float* __restrict__ var,
                           float* __restrict__ x2,
                           int N) {
  int idx = blockIdx.x * blockDim.x + threadIdx.x;
  int total = N * D2;
  if (idx >= total) return;
  int n = idx / D2;
  int o = idx - n * D2;
  float c = fmaxf(cnt[n], 1.0f);
  float h = agg2[idx] / c;
  const float* xr = x1 + (size_t)n * D1;
#pragma unroll
  for (int i = 0; i < D1; ++i) h = fmaf(xr[i], root2[i * D2 + o], h);
  h += bias2[o];
  h = (h - mu[o]) * rsqrtf(var[o] + BN_EPS) * g[o] + be[o];
  x2[idx] = 1.0f / (1.0f + expf(-h));
}

// ------------------------------------------- Gram: out = x2^T @ x2  [160x160]
// fp32 WMMA (V_WMMA_F32_16X16X4_F32). One block per 16x16 output tile,
// 4 waves split K=N; K unrolled x4 with 4 independent accumulators so the
// wmmas are not serialized through the C operand; LDS reduce; wave 0 writes.
// A(16x4): lane<16 -> {K0,K1}, lane>=16 -> {K2,K3}, M=lane%16.
// B(4x16): vgpr0 = row K=2*half, vgpr1 = row K=2*half+1, N=lane%16.
// C/D(16x16, 8 vgprs): M = r + 8*(lane/16), N = lane%16.
__global__ void k_gram_wmma(const float* __restrict__ x2,
                            float* __restrict__ out, int N) {
  const int tile = blockIdx.x;          // 0..99
  const int ti = tile % 10;             // output row tile
  const int tj = tile / 10;             // output col tile
  const int wave = threadIdx.x >> 5;    // 0..3
  const int lane = threadIdx.x & 31;
  const int hf = lane >> 4;             // 0 or 1
  const int lr = lane & 15;
  const int ci = ti * 16 + lr;
  const int cj = tj * 16 + lr;

  const int chunk = N >> 2;             // per-wave K span (4096)
  const int n0 = wave * chunk;
  const int n1 = n0 + chunk;

  v8f acc0 = {0.f,0.f,0.f,0.f,0.f,0.f,0.f,0.f};
  v8f acc1 = acc0, acc2 = acc0, acc3 = acc0;

  for (int n = n0; n < n1; n += 16) {
    const int r0 = n + 2 * hf;
    v2f a, b;
#define GRAM_STEP(ACC, OFF)                                                  \
    a.x = x2[(size_t)(r0 + (OFF)) * D2 + ci];                                \
    a.y = x2[(size_t)(r0 + (OFF) + 1) * D2 + ci];                            \
    b.x = x2[(size_t)(r0 + (OFF)) * D2 + cj];                                \
    b.y = x2[(size_t)(r0 + (OFF) + 1) * D2 + cj];                            \
    ACC = __builtin_amdgcn_wmma_f32_16x16x4_f32(false, a, false, b,          \
                                                (short)0, ACC, false, false);
    GRAM_STEP(acc0, 0)
    GRAM_STEP(acc1, 4)
    GRAM_STEP(acc2, 8)
    GRAM_STEP(acc3, 12)
#undef GRAM_STEP
  }

  v8f acc = (acc0 + acc1) + (acc2 + acc3);

  __shared__ float red[4][32][8];
#pragma unroll
  for (int r = 0; r < 8; ++r) red[wave][lane][r] = acc[r];
  __syncthreads();

  if (wave == 0) {
#pragma unroll
    for (int r = 0; r < 8; ++r) {
      float s = red[0][lane][r] + red[1][lane][r] +
                red[2][lane][r] + red[3][lane][r];
      out[(size_t)(ti * 16 + r + 8 * hf) * D2 + (tj * 16 + lr)] = s;
    }
  }
}

// ---------------------------------------------------------------------- launch
extern "C" void kernel_launch(void* const* d_in, const int* in_sizes, int n_in,
                              void* d_out, int out_size, void* d_ws, size_t ws_size,
                              hipStream_t stream) {
  const float*     x      = (const float*)d_in[0];
  const long long* ei     = (const long long*)d_in[1];   // int64 per reference
  const float*     ea     = (const float*)d_in[2];
  const float*     w1     = (const float*)d_in[3];
  const float*     b1     = (const float*)d_in[4];
  const float*     root1  = (const float*)d_in[5];
  const float*     bias1  = (const float*)d_in[6];
  const float*     bn1g   = (const float*)d_in[7];
  const float*     bn1b   = (const float*)d_in[8];
  const float*     bn1m   = (const float*)d_in[9];
  const float*     bn1v   = (const float*)d_in[10];
  const float*     w2     = (const float*)d_in[11];
  const float*     b2     = (const float*)d_in[12];
  const float*     root2  = (const float*)d_in[13];
  const float*     bias2  = (const float*)d_in[14];
  const float*     bn2g   = (const float*)d_in[15];
  const float*     bn2b   = (const float*)d_in[16];
  const float*     bn2m   = (const float*)d_in[17];
  const float*     bn2v   = (const float*)d_in[18];
  float* out = (float*)d_out;

  const int N = in_sizes[0] / D1;   // 16384
  const int E = in_sizes[2];        // 65536

  // workspace layout (floats): [agg1 N*35][cnt N][agg2 N*160][x1 N*35][x2 N*160]
  float* agg1 = (float*)d_ws;
  float* cnt  = agg1 + (size_t)N * D1;
  float* agg2 = cnt + N;
  float* x1   = agg2 + (size_t)N * D2;
  float* x2   = x1 + (size_t)N * D1;

  const int blk = 256;
  // zero agg1 + cnt + agg2 (contiguous: N*(35+1+160) floats)
  int zn = N * (D1 + 1 + D2);
  k_zero<<<(zn + blk - 1) / blk, blk, 0, stream>>>(agg1, zn);

  int t1 = E * D1;
  k_edge1<<<(t1 + blk - 1) / blk, blk, 0, stream>>>(x, ei, ea, w1, b1, agg1, cnt, E);

  int c1 = N * D1;
  k_combine1<<<(c1 + blk - 1) / blk, blk, 0, stream>>>(x, agg1, cnt, root1, bias1,
                                                       bn1g, bn1b, bn1m, bn1v, x1, N);

  k_edge2<<<(E + EPB - 1) / EPB, D2, 0, stream>>>(x1, ei, ea, w2, b2, agg2, E);

  int c2 = N * D2;
  k_combine2<<<(c2 + blk - 1) / blk, blk, 0, stream>>>(x1, agg2, cnt, root2, bias2,
                                                       bn2g, bn2b, bn2m, bn2v, x2, N);

  // 100 tiles of 16x16, 4 waves each
  k_gram_wmma<<<100, 128, 0, stream>>>(x2, out, N);
}